// ButterworthFilterBank_2808908612109
// MI455X (gfx1250) — compile-verified
//
#include <hip/hip_runtime.h>

// ---------------- sizes / ws layout (in floats) ----------------
#define NSEQ_ALL  1792          // 1536 bank + 256 spec
#define NSEQ_BANK 1536
#define SEGS      64
#define SEGLEN    128
#define TLEN      8192
#define NCFG      7

#define OFF_COEF  0                         // [7][36]   b1,a1,b2,a2
#define OFF_ML    256                       // [7][16][16]  M^128
#define OFF_H     2048                      // [7][128][16] c^T M^t
#define OFF_P     16384                     // [1792][64][16] driven final states
#define OFF_SIG   (16384 + 1792*1024)       // [1792][64][16] segment-start states
#define OFF_MM    (16384 + 2*1792*1024)     // [1536][2] (min, 2/range)

typedef __attribute__((ext_vector_type(2))) float v2f;
typedef __attribute__((ext_vector_type(8))) float v8f;

// ---------------- complex double helpers ----------------
struct cdbl { double re, im; };
__device__ __forceinline__ cdbl cmk(double r, double i) { cdbl z; z.re = r; z.im = i; return z; }
__device__ __forceinline__ cdbl cadd(cdbl a, cdbl b) { return cmk(a.re + b.re, a.im + b.im); }
__device__ __forceinline__ cdbl csub(cdbl a, cdbl b) { return cmk(a.re - b.re, a.im - b.im); }
__device__ __forceinline__ cdbl cmul(cdbl a, cdbl b) { return cmk(a.re*b.re - a.im*b.im, a.re*b.im + a.im*b.re); }
__device__ __forceinline__ cdbl cdivq(cdbl a, cdbl b) {
  double d = b.re*b.re + b.im*b.im;
  return cmk((a.re*b.re + a.im*b.im) / d, (a.im*b.re - a.re*b.im) / d);
}
__device__ __forceinline__ cdbl csqrtd(cdbl z) {
  double r = sqrt(z.re*z.re + z.im*z.im);
  double re = sqrt(fmax(0.0, (r + z.re) * 0.5));
  double im = sqrt(fmax(0.0, (r - z.re) * 0.5));
  if (z.im < 0.0) im = -im;
  return cmk(re, im);
}
__device__ void polyroots(const cdbl* roots, int n, cdbl* c) {
  c[0] = cmk(1.0, 0.0);
  int len = 1;
  for (int k = 0; k < n; ++k) {
    c[len] = cmk(0.0, 0.0);
    for (int j = len; j >= 1; --j) c[j] = csub(c[j], cmul(roots[k], c[j - 1]));
    ++len;
  }
}

// scipy.signal.butter(4, [lo,hi], btype, fs=1000) via zpk + bilinear (double, cast f32)
__device__ void dev_butter(double lo, double hi, int btype /*0 band, 1 stop*/,
                           float* bout, float* aout) {
  const double FS = 1000.0, PI_ = 3.14159265358979323846;
  const int N = 4;
  cdbl pp[4];
  for (int k = 0; k < N; ++k) {
    double m = (double)(-N + 1 + 2 * k);
    double th = PI_ * m / (2.0 * N);
    pp[k] = cmk(-cos(th), -sin(th));              // -exp(i*th)
  }
  double w1 = 2.0 * FS * tan(PI_ * lo / FS);
  double w2 = 2.0 * FS * tan(PI_ * hi / FS);
  double wo = sqrt(w1 * w2), bw = w2 - w1;
  cdbl p2[8], z2[8];
  int nz; double k2;
  if (btype == 0) {
    for (int k = 0; k < N; ++k) {
      cdbl plp = cmk(pp[k].re * bw * 0.5, pp[k].im * bw * 0.5);
      cdbl rad = csqrtd(csub(cmul(plp, plp), cmk(wo * wo, 0.0)));
      p2[k] = cadd(plp, rad); p2[k + N] = csub(plp, rad);
    }
    for (int k = 0; k < N; ++k) z2[k] = cmk(0.0, 0.0);
    nz = N; k2 = bw * bw * bw * bw;
  } else {
    for (int k = 0; k < N; ++k) {
      cdbl php = cdivq(cmk(bw * 0.5, 0.0), pp[k]);
      cdbl rad = csqrtd(csub(cmul(php, php), cmk(wo * wo, 0.0)));
      p2[k] = cadd(php, rad); p2[k + N] = csub(php, rad);
    }
    for (int k = 0; k < N; ++k) { z2[k] = cmk(0.0, wo); z2[k + N] = cmk(0.0, -wo); }
    nz = 2 * N;
    cdbl pr = cmk(1.0, 0.0);
    for (int k = 0; k < N; ++k) pr = cmul(pr, cmk(-pp[k].re, -pp[k].im));
    k2 = cdivq(cmk(1.0, 0.0), pr).re;
  }
  double fs2 = 2.0 * FS;
  cdbl zz[8], pz[8];
  for (int i = 0; i < nz; ++i) zz[i] = cdivq(cadd(cmk(fs2, 0), z2[i]), csub(cmk(fs2, 0), z2[i]));
  for (int i = 0; i < 8;  ++i) pz[i] = cdivq(cadd(cmk(fs2, 0), p2[i]), csub(cmk(fs2, 0), p2[i]));
  for (int i = nz; i < 8; ++i) zz[i] = cmk(-1.0, 0.0);
  cdbl num = cmk(1, 0), den = cmk(1, 0);
  for (int i = 0; i < nz; ++i) num = cmul(num, csub(cmk(fs2, 0), z2[i]));
  for (int i = 0; i < 8;  ++i) den = cmul(den, csub(cmk(fs2, 0), p2[i]));
  double kz = k2 * cdivq(num, den).re;
  cdbl cb[9], ca[9];
  polyroots(zz, 8, cb);
  polyroots(pz, 8, ca);
  for (int i = 0; i < 9; ++i) { bout[i] = (float)(kz * cb[i].re); aout[i] = (float)ca[i].re; }
}

// ---------------- setup: coefficients, M^128, H = c^T M^t ----------------
__global__ void bfb_setup(float* __restrict__ ws) {
  float* coef = ws + OFF_COEF;
  float* ML   = ws + OFF_ML;
  float* H    = ws + OFF_H;
  __shared__ double sM[NCFG][16][16];
  __shared__ double sP[NCFG][16][16];
  __shared__ double sh[NCFG][16];
  __shared__ double shn[NCFG][16];
  int tid = threadIdx.x;

  if (tid < NCFG) {
    const double cuts[7] = {4.0, 8.0, 13.0, 30.0, 45.0, 55.0, 71.0};
    float* b1 = coef + tid * 36; float* a1 = b1 + 9; float* b2 = b1 + 18; float* a2 = b1 + 27;
    double lo, hi;
    if (tid < 6) { lo = cuts[tid]; hi = cuts[tid + 1]; } else { lo = 14.0; hi = 71.0; }
    dev_butter(lo, hi, 0, b1, a1);
    bool notch = (tid == 6) || (lo <= 50.0 && 50.0 <= hi);
    if (notch) dev_butter(48.0, 52.0, 1, b2, a2);
    else for (int i = 0; i < 9; ++i) { b2[i] = (i == 0) ? 1.f : 0.f; a2[i] = (i == 0) ? 1.f : 0.f; }
  }
  __syncthreads();

  // build combined 16-state M for cascade (DF2T filter1 -> filter2), h0 = c
  if (tid < NCFG * 16) {
    int cfg = tid >> 4, row = tid & 15;
    const float* b1 = coef + cfg * 36; const float* a1 = b1 + 9;
    const float* b2 = b1 + 18;         const float* a2 = b1 + 27;
    for (int col = 0; col < 16; ++col) sM[cfg][row][col] = 0.0;
    if (row < 8) {
      int i = row;
      if (i < 7) sM[cfg][row][row + 1] += 1.0;
      sM[cfg][row][0] += -(double)a1[i + 1];
    } else {
      int i = row - 8;
      double g = (double)b2[i + 1] - (double)a2[i + 1] * (double)b2[0];
      if (i < 7) sM[cfg][row][row + 1] += 1.0;
      sM[cfg][row][8] += -(double)a2[i + 1];
      sM[cfg][row][0] += g;
    }
    sh[cfg][row] = (row == 0) ? (double)b2[0] : ((row == 8) ? 1.0 : 0.0);
  }
  __syncthreads();

  // H[cfg][t][j] = (c^T M^t)[j], t = 0..127
  for (int t = 0; t < SEGLEN; ++t) {
    if (tid < NCFG * 16) {
      int cfg = tid >> 4, j = tid & 15;
      H[cfg * (SEGLEN * 16) + t * 16 + j] = (float)sh[cfg][j];
      double acc = 0.0;
      for (int i = 0; i < 16; ++i) acc += sh[cfg][i] * sM[cfg][i][j];
      shn[cfg][j] = acc;
    }
    __syncthreads();
    if (tid < NCFG * 16) { int cfg = tid >> 4, j = tid & 15; sh[cfg][j] = shn[cfg][j]; }
    __syncthreads();
  }

  // M^128 by 7 squarings
  for (int s = 0; s < 7; ++s) {
    if (tid < NCFG * 16) {
      int cfg = tid >> 4, row = tid & 15;
      for (int col = 0; col < 16; ++col) {
        double acc = 0.0;
        for (int k = 0; k < 16; ++k) acc += sM[cfg][row][k] * sM[cfg][k][col];
        sP[cfg][row][col] = acc;
      }
    }
    __syncthreads();
    if (tid < NCFG * 16) {
      int cfg = tid >> 4, row = tid & 15;
      for (int col = 0; col < 16; ++col) sM[cfg][row][col] = sP[cfg][row][col];
    }
    __syncthreads();
  }
  if (tid < NCFG * 16) {
    int cfg = tid >> 4, row = tid & 15;
    for (int col = 0; col < 16; ++col) ML[cfg * 256 + row * 16 + col] = (float)sM[cfg][row][col];
  }
}

// ---------------- pass A: zero-state filtering per (seq, segment) ----------------
__global__ void bfb_passA(const float* __restrict__ x, const float* __restrict__ coef,
                          float* __restrict__ y, float* __restrict__ pstate) {
  int seq = blockIdx.x;
  int seg = threadIdx.x;               // blockDim = 64
  int cfg, xrow;
  if (seq < NSEQ_BANK) { int b = seq / 192, r = seq % 192; cfg = r >> 5; xrow = b * 32 + (r & 31); }
  else { int i = seq - NSEQ_BANK; cfg = 6; xrow = i; }
  const float* cf = coef + cfg * 36;
  float b1[9], a1[9], b2[9], a2[9];
#pragma unroll
  for (int i = 0; i < 9; ++i) { b1[i] = cf[i]; a1[i] = cf[9 + i]; b2[i] = cf[18 + i]; a2[i] = cf[27 + i]; }
  const float* xp = x + (size_t)xrow * TLEN + seg * SEGLEN;
  float*       yp = y + (size_t)seq  * TLEN + seg * SEGLEN;
  float d[8], e[8];
#pragma unroll
  for (int i = 0; i < 8; ++i) { d[i] = 0.f; e[i] = 0.f; }
  for (int t = 0; t < SEGLEN; ++t) {
    float xv = xp[t];
    float y1 = fmaf(b1[0], xv, d[0]);
#pragma unroll
    for (int i = 0; i < 7; ++i) d[i] = fmaf(-a1[i + 1], y1, fmaf(b1[i + 1], xv, d[i + 1]));
    d[7] = fmaf(-a1[8], y1, b1[8] * xv);
    float y2 = fmaf(b2[0], y1, e[0]);
#pragma unroll
    for (int i = 0; i < 7; ++i) e[i] = fmaf(-a2[i + 1], y2, fmaf(b2[i + 1], y1, e[i + 1]));
    e[7] = fmaf(-a2[8], y2, b2[8] * y1);
    yp[t] = y2;
  }
  float* ps = pstate + (size_t)seq * (SEGS * 16) + seg * 16;
#pragma unroll
  for (int i = 0; i < 8; ++i) { ps[i] = d[i]; ps[8 + i] = e[i]; }
}

// ---------------- pass B: scan sigma_k = M^L sigma_{k-1} + p_{k-1} (16 lanes/seq) ----
__global__ void bfb_passB(const float* __restrict__ ML, const float* __restrict__ pstate,
                          float* __restrict__ sig) {
  int g = blockIdx.x * blockDim.x + threadIdx.x;   // blockDim = 128
  int seq = g >> 4, j = g & 15;
  int cfg = (seq < NSEQ_BANK) ? ((seq % 192) >> 5) : 6;
  const float* mr = ML + cfg * 256 + j * 16;
  float ml[16];
#pragma unroll
  for (int t = 0; t < 16; ++t) ml[t] = mr[t];
  float s = 0.f;
  sig[(size_t)seq * (SEGS * 16) + j] = 0.f;        // sigma_0 = 0
  for (int k = 1; k < SEGS; ++k) {
    float ns = pstate[(size_t)seq * (SEGS * 16) + (k - 1) * 16 + j];
#pragma unroll
    for (int t = 0; t < 16; ++t) ns = fmaf(ml[t], __shfl(s, t, 16), ns);
    s = ns;
    sig[(size_t)seq * (SEGS * 16) + k * 16 + j] = s;
  }
}

// ---------------- pass C: WMMA correction  y += H(128x16) * Sigma(16x16 seqs) --------
// A 16x4 f32 layout: lanes 0-15 -> rows M, v0:K={0|2}, v1:K={1|3} (ISA 7.12.2).
// B 4x16 assumed mirrored: lanes -> cols N, v0:K={0|2}, v1:K={1|3}.
// C/D: VGPR r -> M=r (lanes 0-15) / M=r+8 (lanes 16-31), N=lane%16.
__global__ void bfb_passC(const float* __restrict__ Hc, const float* __restrict__ sig,
                          float* __restrict__ y) {
  int gid = blockIdx.x;                 // grp*64 + seg, blockDim = 32 (one wave)
  int grp = gid >> 6, seg = gid & 63;
  int seq0, cfg;
  if (grp < 96) { int b = grp / 12, r = grp % 12; cfg = r >> 1; seq0 = (b * 6 + cfg) * 32 + (r & 1) * 16; }
  else          { int i = grp - 96;    cfg = 6;   seq0 = NSEQ_BANK + (i >> 1) * 32 + (i & 1) * 16; }
  int lane = threadIdx.x;
  int ln = lane & 15;
  int hi2 = (lane >> 4) << 1;           // 0 or 2
  const float* Hf = Hc + cfg * (SEGLEN * 16);

  // B fragments (Sigma), reused by all 8 output tiles
  v2f bfr[4];
#pragma unroll
  for (int kk = 0; kk < 4; ++kk) {
    const float* sp = sig + (size_t)(seq0 + ln) * (SEGS * 16) + seg * 16 + kk * 4 + hi2;
    bfr[kk].x = sp[0]; bfr[kk].y = sp[1];
  }
#pragma unroll
  for (int tt = 0; tt < 8; ++tt) {
    int t0 = tt << 4;
    float* yb = y + (size_t)(seq0 + ln) * TLEN + seg * SEGLEN + t0 + (hi2 << 2);
    // hoist all 4 A fragments so they issue as one load clause (single wait),
    // letting the 4-deep WMMA accumulation chain run back-to-back
    v2f afr[4];
#pragma unroll
    for (int kk = 0; kk < 4; ++kk) {
      const float* hp = Hf + (t0 + ln) * 16 + kk * 4 + hi2;
      afr[kk].x = hp[0]; afr[kk].y = hp[1];
    }
    v8f c;
#pragma unroll
    for (int r = 0; r < 8; ++r) c[r] = yb[r];
#pragma unroll
    for (int kk = 0; kk < 4; ++kk) {
      c = __builtin_amdgcn_wmma_f32_16x16x4_f32(false, afr[kk], false, bfr[kk],
                                                (short)0, c, false, false);
    }
#pragma unroll
    for (int r = 0; r < 8; ++r) yb[r] = c[r];
  }
}

// ---------------- pass D: per-bank-sequence min / 2/range ----------------
__global__ void bfb_passD(const float* __restrict__ y, float* __restrict__ mm) {
  __shared__ float smn[256], smx[256];
  int seq = blockIdx.x;
  const float* yp = y + (size_t)seq * TLEN;
  float mn = 3.402823466e38f, mx = -3.402823466e38f;
  for (int t = threadIdx.x; t < TLEN; t += 256) {
    float v = yp[t];
    mn = fminf(mn, v); mx = fmaxf(mx, v);
  }
  smn[threadIdx.x] = mn; smx[threadIdx.x] = mx;
  __syncthreads();
  for (int s = 128; s > 0; s >>= 1) {
    if (threadIdx.x < s) {
      smn[threadIdx.x] = fminf(smn[threadIdx.x], smn[threadIdx.x + s]);
      smx[threadIdx.x] = fmaxf(smx[threadIdx.x], smx[threadIdx.x + s]);
    }
    __syncthreads();
  }
  if (threadIdx.x == 0) {
    float mnv = smn[0];
    mm[seq * 2] = mnv;
    mm[seq * 2 + 1] = 2.0f / (smx[0] - mnv);
  }
}

// ---------------- pass E: normalize bank region to [-1, 1] ----------------
__global__ void bfb_passE(float* __restrict__ y, const float* __restrict__ mm) {
  size_t idx = (size_t)blockIdx.x * blockDim.x + threadIdx.x;  // exactly 1536*8192
  int seq = (int)(idx >> 13);
  float mn = mm[seq * 2], sc = mm[seq * 2 + 1];
  y[idx] = fmaf(y[idx] - mn, sc, -1.0f);
}

extern "C" void kernel_launch(void* const* d_in, const int* in_sizes, int n_in,
                              void* d_out, int out_size, void* d_ws, size_t ws_size,
                              hipStream_t stream) {
  (void)in_sizes; (void)n_in; (void)out_size; (void)ws_size;
  const float* x = (const float*)d_in[0];
  float* out = (float*)d_out;
  float* ws  = (float*)d_ws;
  float* coef = ws + OFF_COEF;
  float* ML   = ws + OFF_ML;
  float* H    = ws + OFF_H;
  float* p    = ws + OFF_P;
  float* sig  = ws + OFF_SIG;
  float* mm   = ws + OFF_MM;

  bfb_setup<<<1, 256, 0, stream>>>(ws);
  bfb_passA<<<NSEQ_ALL, SEGS, 0, stream>>>(x, coef, out, p);
  bfb_passB<<<NSEQ_ALL * 16 / 128, 128, 0, stream>>>(ML, p, sig);
  bfb_passC<<<112 * SEGS, 32, 0, stream>>>(H, sig, out);
  bfb_passD<<<NSEQ_BANK, 256, 0, stream>>>(out, mm);
  bfb_passE<<<NSEQ_BANK * TLEN / 256, 256, 0, stream>>>(out, mm);
}